// GATNet_57793079935362
// MI455X (gfx1250) — compile-verified
//
#include <hip/hip_runtime.h>
#include <hip/hip_bf16.h>

#define NN 100000
#define NE 1600000
#define F_IN 128
#define H1 4
#define D1 32
#define F_MID 128   /* H1*D1 */
#define D2 64
#define NEG_SLOPE 0.2f
#define ENC_NEG_INF 0x007FFFFFu

typedef float v2f __attribute__((ext_vector_type(2)));
typedef float v8f __attribute__((ext_vector_type(8)));

// ---------- monotonic uint encoding for float atomic max ----------
__device__ __forceinline__ unsigned fenc(float f) {
    unsigned u = __float_as_uint(f);
    return (u & 0x80000000u) ? ~u : (u | 0x80000000u);
}
__device__ __forceinline__ float fdec(unsigned u) {
    return (u & 0x80000000u) ? __uint_as_float(u & 0x7FFFFFFFu)
                             : __uint_as_float(~u);
}

// ---------- generic fills ----------
__global__ void fill_u32(unsigned* __restrict__ p, unsigned v, long long n) {
    long long i = (long long)blockIdx.x * blockDim.x + threadIdx.x;
    if (i < n) p[i] = v;
}
__global__ void fill_zero_v4(float4* __restrict__ p, long long n4) {
    long long i = (long long)blockIdx.x * blockDim.x + threadIdx.x;
    if (i < n4) p[i] = make_float4(0.f, 0.f, 0.f, 0.f);
}

// ---------- fp32 WMMA GEMM: C[M,N] = A[M,K] @ B[K,N] ----------
// One wave computes a 16x16 tile with V_WMMA_F32_16X16X4_F32.
// M must be a multiple of 16 (100000 = 6250*16), N multiple of 16, K of 4.
__global__ void __launch_bounds__(128)
gemm_f32_wmma(const float* __restrict__ A, const float* __restrict__ B,
              float* __restrict__ C, int M, int N, int K) {
    int lane  = threadIdx.x;                       // 0..31 (wave32)
    int ntile = blockIdx.x * blockDim.y + threadIdx.y;
    int mtile = blockIdx.y;
    if (ntile * 16 >= N) return;                   // uniform per wave
    int half = lane >> 4;                          // 0/1 -> K-half
    int idx  = lane & 15;                          // row (A) / col (B,C)

    const float* Arow = A + (size_t)(mtile * 16 + idx) * K;
    int col = ntile * 16 + idx;

    v8f acc = (v8f){0.f, 0.f, 0.f, 0.f, 0.f, 0.f, 0.f, 0.f};
    for (int k = 0; k < K; k += 4) {
        int ka = k + half * 2;
        v2f a, b;
        a.x = Arow[ka];
        a.y = Arow[ka + 1];
        b.x = B[(size_t)ka * N + col];
        b.y = B[(size_t)(ka + 1) * N + col];
        acc = __builtin_amdgcn_wmma_f32_16x16x4_f32(
            /*neg_a=*/false, a, /*neg_b=*/false, b,
            /*c_mod=*/(short)0, acc, /*reuse_a=*/false, /*reuse_b=*/false);
    }
    int rowbase = mtile * 16 + half * 8;
#pragma unroll
    for (int v = 0; v < 8; ++v)
        C[(size_t)(rowbase + v) * N + col] = acc[v];
}

// ---------- per-(node,head) attention terms ----------
__global__ void attn_terms(const float* __restrict__ feat,
                           const float* __restrict__ attn_l,
                           const float* __restrict__ attn_r,
                           float* __restrict__ el, float* __restrict__ er,
                           int n, int H, int D) {
    int i = blockIdx.x * blockDim.x + threadIdx.x;   // over n*H
    if (i >= n * H) return;
    int nn = i / H, h = i % H;
    const float* f  = feat + ((size_t)nn * H + h) * D;
    const float* al = attn_l + (size_t)h * D;
    const float* ar = attn_r + (size_t)h * D;
    float sl = 0.f, sr = 0.f;
    for (int d = 0; d < D; ++d) {
        float v = f[d];
        sl += v * al[d];
        sr += v * ar[d];
    }
    el[i] = sl;
    er[i] = sr;
}

// ---------- edge pass 1: leaky-relu logits + segment max ----------
__global__ void edge_logits(const int* __restrict__ src, const int* __restrict__ dst,
                            const float* __restrict__ el, const float* __restrict__ er,
                            float* __restrict__ ebuf, unsigned* __restrict__ menc, int H) {
    long long i = (long long)blockIdx.x * blockDim.x + threadIdx.x;  // over E*H
    if (i >= (long long)NE * H) return;
    int e = (int)(i / H), h = (int)(i % H);
    int s = src[e], d = dst[e];
    float v = el[s * H + h] + er[d * H + h];
    v = (v > 0.f) ? v : NEG_SLOPE * v;
    ebuf[i] = v;
    atomicMax(&menc[d * H + h], fenc(v));
}

// ---------- edge pass 2: exp(e - m) + segment sum ----------
__global__ void edge_expsum(const int* __restrict__ dst,
                            float* __restrict__ ebuf,
                            const unsigned* __restrict__ menc,
                            float* __restrict__ den, int H) {
    long long i = (long long)blockIdx.x * blockDim.x + threadIdx.x;  // over E*H
    if (i >= (long long)NE * H) return;
    int e = (int)(i / H), h = (int)(i % H);
    int d = dst[e];
    float m  = fdec(menc[d * H + h]);
    float ex = expf(ebuf[i] - m);
    ebuf[i] = ex;
    atomicAdd(&den[d * H + h], ex);
}

// ---------- edge pass 2.5: normalize -> alpha in place ----------
__global__ void edge_alpha(const int* __restrict__ dst,
                           float* __restrict__ ebuf,
                           const float* __restrict__ den, int H) {
    long long i = (long long)blockIdx.x * blockDim.x + threadIdx.x;  // over E*H
    if (i >= (long long)NE * H) return;
    int e = (int)(i / H), h = (int)(i % H);
    ebuf[i] = ebuf[i] / den[dst[e] * H + h];
}

// ---------- edge pass 3: scatter messages (float4 per thread) ----------
__global__ void edge_scatter(const int* __restrict__ src, const int* __restrict__ dst,
                             const float* __restrict__ feat,
                             const float* __restrict__ alpha,
                             float* __restrict__ out, int H, int Dv /* = D/4 */) {
    long long i = (long long)blockIdx.x * blockDim.x + threadIdx.x;  // over E*H*Dv
    if (i >= (long long)NE * H * Dv) return;
    int q = (int)(i % Dv);
    long long t = i / Dv;
    int h = (int)(t % H);
    long long e = t / H;
    int s = src[e], dd = dst[e];
    float a = alpha[e * H + h];
    const float4 f = ((const float4*)(feat + ((size_t)s * H + h) * (Dv * 4)))[q];
    float* o = out + ((size_t)dd * H + h) * (Dv * 4) + (size_t)q * 4;
    atomicAdd(o + 0, f.x * a);
    atomicAdd(o + 1, f.y * a);
    atomicAdd(o + 2, f.z * a);
    atomicAdd(o + 3, f.w * a);
}

// ---------- bias + ELU (in place) ----------
__global__ void bias_elu(float* __restrict__ x, const float* __restrict__ bias,
                         long long n, int F) {
    long long i = (long long)blockIdx.x * blockDim.x + threadIdx.x;
    if (i >= n) return;
    float v = x[i] + bias[(int)(i % F)];
    x[i] = (v > 0.f) ? v : expm1f(v);
}

// ---------- bias add (in place) ----------
__global__ void bias_add(float* __restrict__ x, const float* __restrict__ bias,
                         long long n, int F) {
    long long i = (long long)blockIdx.x * blockDim.x + threadIdx.x;
    if (i >= n) return;
    x[i] = x[i] + bias[(int)(i % F)];
}

static inline int blks(long long n, int b) { return (int)((n + b - 1) / b); }

extern "C" void kernel_launch(void* const* d_in, const int* in_sizes, int n_in,
                              void* d_out, int out_size, void* d_ws, size_t ws_size,
                              hipStream_t stream) {
    (void)in_sizes; (void)n_in; (void)out_size; (void)ws_size;

    const float* features = (const float*)d_in[0];   // [N, 128]
    const int*   src      = (const int*)d_in[1];     // [E]
    const int*   dst      = (const int*)d_in[2];     // [E]
    const float* fc1_w    = (const float*)d_in[3];   // [128, 128]
    const float* attn_l1  = (const float*)d_in[4];   // [4, 32]
    const float* attn_r1  = (const float*)d_in[5];   // [4, 32]
    const float* bias1    = (const float*)d_in[6];   // [128]
    const float* fc2_w    = (const float*)d_in[7];   // [128, 64]
    const float* attn_l2  = (const float*)d_in[8];   // [1, 64]
    const float* attn_r2  = (const float*)d_in[9];   // [1, 64]
    const float* bias2    = (const float*)d_in[10];  // [64]
    float* out = (float*)d_out;                      // [N, 64]

    // ---- workspace layout (floats) ----
    float* ws = (float*)d_ws;
    size_t o = 0;
    float*    feat1 = ws + o; o += (size_t)NN * F_MID;   // [N,128], later reused as feat2
    float*    out1  = ws + o; o += (size_t)NN * F_MID;   // [N,128] layer-1 output / x2
    float*    ebuf  = ws + o; o += (size_t)NE * H1;      // [E,4], shared with layer 2
    float*    el1   = ws + o; o += (size_t)NN * H1;
    float*    er1   = ws + o; o += (size_t)NN * H1;
    float*    den1  = ws + o; o += (size_t)NN * H1;
    unsigned* menc1 = (unsigned*)(ws + o); o += (size_t)NN * H1;
    float*    el2   = ws + o; o += (size_t)NN;
    float*    er2   = ws + o; o += (size_t)NN;
    float*    den2  = ws + o; o += (size_t)NN;
    unsigned* menc2 = (unsigned*)(ws + o); o += (size_t)NN;
    float*    feat2 = feat1;  // alias: feat1 dead after layer-1 scatter

    const int TB = 256;

    // ---- init accumulators / softmax state ----
    fill_zero_v4<<<blks((long long)NN * F_MID / 4, TB), TB, 0, stream>>>((float4*)out1, (long long)NN * F_MID / 4);
    fill_zero_v4<<<blks((long long)NN * D2 / 4,   TB), TB, 0, stream>>>((float4*)out,  (long long)NN * D2 / 4);
    fill_u32<<<blks((long long)NN * H1, TB), TB, 0, stream>>>((unsigned*)den1, 0u, (long long)NN * H1);
    fill_u32<<<blks((long long)NN * H1, TB), TB, 0, stream>>>(menc1, ENC_NEG_INF, (long long)NN * H1);
    fill_u32<<<blks((long long)NN,      TB), TB, 0, stream>>>((unsigned*)den2, 0u, (long long)NN);
    fill_u32<<<blks((long long)NN,      TB), TB, 0, stream>>>(menc2, ENC_NEG_INF, (long long)NN);

    // ================= Layer 1 =================
    {
        dim3 blk(32, 4);
        dim3 grd((F_MID / 16 + 3) / 4, NN / 16);   // (2, 6250)
        gemm_f32_wmma<<<grd, blk, 0, stream>>>(features, fc1_w, feat1, NN, F_MID, F_IN);
    }
    attn_terms<<<blks((long long)NN * H1, TB), TB, 0, stream>>>(feat1, attn_l1, attn_r1, el1, er1, NN, H1, D1);
    edge_logits<<<blks((long long)NE * H1, TB), TB, 0, stream>>>(src, dst, el1, er1, ebuf, menc1, H1);
    edge_expsum<<<blks((long long)NE * H1, TB), TB, 0, stream>>>(dst, ebuf, menc1, den1, H1);
    edge_alpha <<<blks((long long)NE * H1, TB), TB, 0, stream>>>(dst, ebuf, den1, H1);
    edge_scatter<<<blks((long long)NE * H1 * (D1 / 4), TB), TB, 0, stream>>>(src, dst, feat1, ebuf, out1, H1, D1 / 4);
    bias_elu<<<blks((long long)NN * F_MID, TB), TB, 0, stream>>>(out1, bias1, (long long)NN * F_MID, F_MID);

    // ================= Layer 2 =================
    {
        dim3 blk(32, 4);
        dim3 grd((D2 / 16 + 3) / 4, NN / 16);      // (1, 6250)
        gemm_f32_wmma<<<grd, blk, 0, stream>>>(out1, fc2_w, feat2, NN, D2, F_MID);
    }
    attn_terms<<<blks((long long)NN, TB), TB, 0, stream>>>(feat2, attn_l2, attn_r2, el2, er2, NN, 1, D2);
    edge_logits<<<blks((long long)NE, TB), TB, 0, stream>>>(src, dst, el2, er2, ebuf, menc2, 1);
    edge_expsum<<<blks((long long)NE, TB), TB, 0, stream>>>(dst, ebuf, menc2, den2, 1);
    edge_alpha <<<blks((long long)NE, TB), TB, 0, stream>>>(dst, ebuf, den2, 1);
    edge_scatter<<<blks((long long)NE * (D2 / 4), TB), TB, 0, stream>>>(src, dst, feat2, ebuf, out, 1, D2 / 4);
    bias_add<<<blks((long long)NN * D2, TB), TB, 0, stream>>>(out, bias2, (long long)NN * D2, D2);
}